// Target_GraphSAGE_88493506166796
// MI455X (gfx1250) — compile-verified
//
#include <hip/hip_runtime.h>
#include <hip/hip_bf16.h>

#define N_NODES_C 100000
#define N_EDGES_C 1600000
#define FEAT_C    128

typedef float v2f __attribute__((ext_vector_type(2)));
typedef float v8f __attribute__((ext_vector_type(8)));

// ---------------------------------------------------------------- zero fill
__global__ void zero_f32(float* __restrict__ p, size_t n) {
  size_t i = (size_t)blockIdx.x * blockDim.x + threadIdx.x;
  size_t stride = (size_t)gridDim.x * blockDim.x;
  for (; i < n; i += stride) p[i] = 0.0f;
}

// ------------------------------------------------- edge scatter (mean agg)
// One wave32 per edge: lane l moves features [4l, 4l+3] (128 = 32*4).
__global__ __launch_bounds__(256) void sage_scatter(
    const float* __restrict__ feat, const long long* __restrict__ src,
    const long long* __restrict__ dst, float* __restrict__ agg,
    float* __restrict__ cnt, int nEdges) {
  int wave = (int)((blockIdx.x * blockDim.x + threadIdx.x) >> 5);
  int lane = (int)(threadIdx.x & 31u);
  if (wave >= nEdges) return;
  int s = (int)src[wave];
  int d = (int)dst[wave];
  const float4 v = *(const float4*)(feat + (size_t)s * FEAT_C + lane * 4);
  float* dp = agg + (size_t)d * FEAT_C + lane * 4;
  atomicAdd(dp + 0, v.x);
  atomicAdd(dp + 1, v.y);
  atomicAdd(dp + 2, v.z);
  atomicAdd(dp + 3, v.w);
  if (lane == 0) atomicAdd(cnt + d, 1.0f);
}

// ---------------------------------------- fused SAGE linear via f32 WMMA
// out[m, n] = sum_k mean[m,k]*Wl[n,k] + bl[n] + sum_k x[m,k]*Wr[n,k]
// mean[m,k] = agg[m,k] / max(cnt[m], 1)
// One wave per 16-row tile; NOUT/16 column tiles of 16x16 accumulated with
// V_WMMA_F32_16X16X4_F32 (K=4 per instruction).
template <int NOUT, bool RELU>
__global__ __launch_bounds__(256) void sage_gemm(
    const float* __restrict__ xin,   // N x 128
    const float* __restrict__ agg,   // N x 128
    const float* __restrict__ cnt,   // N
    const float* __restrict__ Wl,    // NOUT x 128
    const float* __restrict__ bl,    // NOUT
    const float* __restrict__ Wr,    // NOUT x 128
    float* __restrict__ out,         // N x NOUT
    int nNodes) {
  const int wave = (int)((blockIdx.x * blockDim.x + threadIdx.x) >> 5);
  const int lane = (int)(threadIdx.x & 31u);
  const int m0 = wave * 16;
  if (m0 >= nNodes) return;  // wave-uniform: EXEC stays all-ones for WMMA

  // A fragment addressing (f32 16x4): lane%16 = M row, (lane/16)*2 = K base
  const int mA = m0 + (lane & 15);
  const int kk = (lane >> 4) << 1;           // 0 or 2
  const int nB = (lane & 15);                // B fragment: lane%16 = N col

  const float inv = 1.0f / fmaxf(cnt[mA], 1.0f);
  const float* arow = agg + (size_t)mA * FEAT_C + kk;
  const float* xrow = xin + (size_t)mA * FEAT_C + kk;

  constexpr int NT = NOUT / 16;
  v8f acc[NT] = {};

  // Pass 1: mean (= agg * inv) against Wl
  for (int k0 = 0; k0 < FEAT_C; k0 += 4) {
    v2f a;
    a.x = arow[k0] * inv;
    a.y = arow[k0 + 1] * inv;
#pragma unroll
    for (int t = 0; t < NT; ++t) {
      const float* wrow = Wl + (size_t)(t * 16 + nB) * FEAT_C + kk + k0;
      v2f b;
      b.x = wrow[0];
      b.y = wrow[1];
      acc[t] = __builtin_amdgcn_wmma_f32_16x16x4_f32(
          false, a, false, b, (short)0, acc[t], false, false);
    }
  }
  // Pass 2: x against Wr
  for (int k0 = 0; k0 < FEAT_C; k0 += 4) {
    v2f a;
    a.x = xrow[k0];
    a.y = xrow[k0 + 1];
#pragma unroll
    for (int t = 0; t < NT; ++t) {
      const float* wrow = Wr + (size_t)(t * 16 + nB) * FEAT_C + kk + k0;
      v2f b;
      b.x = wrow[0];
      b.y = wrow[1];
      acc[t] = __builtin_amdgcn_wmma_f32_16x16x4_f32(
          false, a, false, b, (short)0, acc[t], false, false);
    }
  }

  // Epilogue: C/D layout -> vgpr r holds row m0+r (lanes 0-15) / m0+r+8
  const int halfOff = (lane >> 4) * 8;
#pragma unroll
  for (int t = 0; t < NT; ++t) {
    const float bias = bl[t * 16 + nB];
#pragma unroll
    for (int r = 0; r < 8; ++r) {
      const int m = m0 + r + halfOff;
      float v = acc[t][r] + bias;
      if (RELU) v = fmaxf(v, 0.0f);
      out[(size_t)m * NOUT + t * 16 + nB] = v;
    }
  }
}

// --------------------------------------------- log_softmax over 64 classes
// One wave per node; 2 classes per lane; wave32 shfl_xor reductions.
__global__ __launch_bounds__(256) void logsoftmax64(float* __restrict__ out,
                                                    int nNodes) {
  int wave = (int)((blockIdx.x * blockDim.x + threadIdx.x) >> 5);
  int lane = (int)(threadIdx.x & 31u);
  if (wave >= nNodes) return;
  float* row = out + (size_t)wave * 64;
  float2 v = *(float2*)(row + lane * 2);
  float mx = fmaxf(v.x, v.y);
  for (int off = 16; off > 0; off >>= 1)
    mx = fmaxf(mx, __shfl_xor(mx, off, 32));
  float s = expf(v.x - mx) + expf(v.y - mx);
  for (int off = 16; off > 0; off >>= 1)
    s += __shfl_xor(s, off, 32);
  const float lse = mx + logf(s);
  float2 o;
  o.x = v.x - lse;
  o.y = v.y - lse;
  *(float2*)(row + lane * 2) = o;
}

extern "C" void kernel_launch(void* const* d_in, const int* in_sizes, int n_in,
                              void* d_out, int out_size, void* d_ws,
                              size_t ws_size, hipStream_t stream) {
  const float* x = (const float*)d_in[0];
  const long long* ei = (const long long*)d_in[1];  // int64 per reference
  const float* W1l = (const float*)d_in[2];
  const float* b1l = (const float*)d_in[3];
  const float* W1r = (const float*)d_in[4];
  const float* W2l = (const float*)d_in[5];
  const float* b2l = (const float*)d_in[6];
  const float* W2r = (const float*)d_in[7];
  float* out = (float*)d_out;

  const long long* src = ei;
  const long long* dst = ei + N_EDGES_C;

  float* agg = (float*)d_ws;                         // N*128 f32
  float* cnt = agg + (size_t)N_NODES_C * FEAT_C;     // N f32
  float* h = cnt + N_NODES_C;                        // N*128 f32
  const size_t aggcnt = (size_t)N_NODES_C * FEAT_C + N_NODES_C;

  const int scatterBlocks = N_EDGES_C / 8;              // 8 waves/block
  const int gemmBlocks = (N_NODES_C / 16 + 7) / 8;      // 16 rows/wave
  const int lsmBlocks = (N_NODES_C + 7) / 8;            // 1 wave/node

  // ---- layer 1
  zero_f32<<<2048, 256, 0, stream>>>(agg, aggcnt);
  sage_scatter<<<scatterBlocks, 256, 0, stream>>>(x, src, dst, agg, cnt,
                                                  N_EDGES_C);
  sage_gemm<128, true><<<gemmBlocks, 256, 0, stream>>>(x, agg, cnt, W1l, b1l,
                                                       W1r, h, N_NODES_C);
  // ---- layer 2
  zero_f32<<<2048, 256, 0, stream>>>(agg, aggcnt);
  sage_scatter<<<scatterBlocks, 256, 0, stream>>>(h, src, dst, agg, cnt,
                                                  N_EDGES_C);
  sage_gemm<64, false><<<gemmBlocks, 256, 0, stream>>>(h, agg, cnt, W2l, b2l,
                                                       W2r, out, N_NODES_C);
  logsoftmax64<<<lsmBlocks, 256, 0, stream>>>(out, N_NODES_C);
}